// DeformConv1d_6373731467652
// MI455X (gfx1250) — compile-verified
//
#include <hip/hip_runtime.h>

typedef float v2f __attribute__((ext_vector_type(2)));
typedef float v8f __attribute__((ext_vector_type(8)));

#define B_SZ   8
#define C_IN   64
#define L_IN   16384
#define C_OUT  64
#define KS     5
#define L_OUT  (L_IN - (KS - 1))     // 16380
#define CK     (C_IN * KS)           // 320
#define LT     32                    // l-tile per block
#define NTILES ((L_OUT + LT - 1) / LT)  // 512 (power of two -> shift)
#define WPITCH 65                    // lds weight pitch (floats), [CK][WPITCH]
#define XPITCH (CK + 2)              // 322, xoff [LT][XPITCH], 8B-aligned rows

struct SharedMem {
    float w[CK * WPITCH];       // 83200 B : weight transposed [ck][o]
    float xoff[LT * XPITCH];    // 41216 B : interpolated samples [l][ck]
    int   i0[LT * KS];
    int   i1[LT * KS];
    float w0[LT * KS];
    float w1[LT * KS];
};                              // ~124 KB total (gfx1250: 320 KB/WGP)

__global__ __launch_bounds__(256) void deform_conv1d_wmma(
    const float* __restrict__ x,      // [B, C_IN, L_IN]
    const float* __restrict__ off,    // [B, 1, L_OUT, K]
    const float* __restrict__ wt,     // [C_OUT, C_IN, K]
    const float* __restrict__ bias,   // [C_OUT]
    float* __restrict__ out)          // [B, C_OUT, L_OUT]
{
    __shared__ SharedMem sm;
    const int tid  = threadIdx.x;
    const int lane = tid & 31;
    const int wave = tid >> 5;

    const int tile = blockIdx.x;      // b*512 + l_tile
    const int b    = tile >> 9;       // 512 tiles per batch
    const int l0   = (tile & (NTILES - 1)) * LT;

    // ---- stage 0a: weight -> LDS, transposed [ck][o], pitch 65 ----
    for (int e = tid; e < C_OUT * CK; e += 256) {
        int o  = e / CK;
        int ck = e - o * CK;
        sm.w[ck * WPITCH + o] = wt[e];      // coalesced read, conflict-free write
    }

    // ---- stage 0b: per-(l,k) interpolation state ----
    if (tid < LT * KS) {
        int l  = tid / KS;
        int k  = tid - l * KS;
        int lg = l0 + l;
        float w0 = 0.f, w1 = 0.f;
        int   i0 = 0,   i1 = 0;
        if (lg < L_OUT) {
            float T   = (float)lg + (float)k +
                        off[(size_t)(b * L_OUT + lg) * KS + k];
            float fl  = floorf(T);
            int   ii0 = (int)fl;
            int   ii1 = ii0 + 1;
            float fr  = T - fl;
            w0 = (ii0 >= 0 && ii0 <= L_IN - 1) ? (1.f - fr) : 0.f;
            w1 = (ii1 >= 0 && ii1 <= L_IN - 1) ? fr : 0.f;
            i0 = min(max(ii0, 0), L_IN - 1);
            i1 = min(max(ii1, 0), L_IN - 1);
        }
        sm.i0[tid] = i0; sm.i1[tid] = i1;
        sm.w0[tid] = w0; sm.w1[tid] = w1;
    }
    __syncthreads();

    // ---- stage 1: gather + bilinear interp into xoff[l][ck] ----
    {
        const float* xb = x + (size_t)b * C_IN * L_IN;
        const int l  = tid & 31;       // lane-contiguous l -> near-coalesced gathers
        const int q0 = tid >> 5;       // starting ck
        #pragma unroll
        for (int it = 0; it < CK / 8; ++it) {       // 40 iters
            int q = q0 + it * 8;                    // ck = c*KS + k
            int c = q / KS;
            int k = q - c * KS;
            int p = l * KS + k;
            float a0 = xb[(size_t)c * L_IN + sm.i0[p]];
            float a1 = xb[(size_t)c * L_IN + sm.i1[p]];
            sm.xoff[l * XPITCH + q] = a0 * sm.w0[p] + a1 * sm.w1[p];
        }
    }
    __syncthreads();

    // ---- stage 2: per-wave 16x16 tile, 80 x v_wmma_f32_16x16x4_f32 ----
    const int ob     = (wave & 3) * 16;   // output-channel base
    const int lb     = (wave >> 2) * 16;  // l base within tile
    const int lane16 = lane & 15;
    const int klo    = (lane >> 4) * 2;   // lanes 0-15: K{0,1}; 16-31: K{2,3}

    v8f acc = {};
    #pragma unroll 8
    for (int kk = 0; kk < CK; kk += 4) {
        v2f a, bm;
        // A (16x4): A[M=lane16][kk+klo], [kk+klo+1] from transposed weight
        a.x = sm.w[(kk + klo)     * WPITCH + ob + lane16];
        a.y = sm.w[(kk + klo + 1) * WPITCH + ob + lane16];
        // B (4x16): B[kk+klo..+1][N=lane16] -> contiguous pair in [l][ck]
        bm = *(const v2f*)&sm.xoff[(lb + lane16) * XPITCH + kk + klo];
        acc = __builtin_amdgcn_wmma_f32_16x16x4_f32(
                  false, a, false, bm, (short)0, acc, false, false);
    }

    // ---- epilogue: C/D layout -> rows M=r (lanes 0-15) / M=r+8 (lanes 16-31)
    const int lg = l0 + lb + lane16;
    if (lg < L_OUT) {
        const int half = lane >> 4;
        #pragma unroll
        for (int r = 0; r < 8; ++r) {
            int o = ob + r + half * 8;
            out[((size_t)b * C_OUT + o) * L_OUT + lg] = acc[r] + bias[o];
        }
    }
}

extern "C" void kernel_launch(void* const* d_in, const int* in_sizes, int n_in,
                              void* d_out, int out_size, void* d_ws, size_t ws_size,
                              hipStream_t stream) {
    const float* x    = (const float*)d_in[0];
    const float* off  = (const float*)d_in[1];
    const float* wt   = (const float*)d_in[2];
    const float* bias = (const float*)d_in[3];
    // d_in[4..6] = kernel_size, dilation, stride (compile-time constants here)
    float* out = (float*)d_out;

    dim3 grid(B_SZ * NTILES);   // 8 * 512 = 4096 blocks
    dim3 block(256);            // 8 waves (wave32)
    deform_conv1d_wmma<<<grid, block, 0, stream>>>(x, off, wt, bias, out);
}